// KNN_Pooling_50749333570130
// MI455X (gfx1250) — compile-verified
//
#include <hip/hip_runtime.h>
#include <hip/hip_bf16.h>

typedef float v2f __attribute__((ext_vector_type(2)));
typedef float v8f __attribute__((ext_vector_type(8)));

#define NPTS   4096
#define NSAMP  1024
#define NCH    128
#define NBATCH 16
#define KNN    32

#define FLT_BIG 3.402823466e38f

// ---------------------------------------------------------------------------
// Kernel A: farthest point sampling. One 256-thread workgroup per batch.
// Coordinates held in registers (16 pts/thread); LDS for argmax tree +
// broadcast of the current point. Writes sampled coordinates [B,S,3].
// ---------------------------------------------------------------------------
__global__ void fps_kernel(const float* __restrict__ coords,
                           float* __restrict__ samp_xyz) {
  const int b = blockIdx.x;
  const int t = threadIdx.x;
  const float* cb = coords + (size_t)b * NPTS * 3;

  float px[16], py[16], pz[16], md[16];
#pragma unroll
  for (int j = 0; j < 16; ++j) {
    int n = j * 256 + t;
    px[j] = cb[n * 3 + 0];
    py[j] = cb[n * 3 + 1];
    pz[j] = cb[n * 3 + 2];
    md[j] = FLT_BIG;
  }

  __shared__ float s_val[256];
  __shared__ int   s_idx[256];
  __shared__ float s_cx, s_cy, s_cz;

  int cur = 0;
  for (int it = 0; it < NSAMP; ++it) {
    // broadcast coordinates of 'cur' from the owning thread
    if (t == (cur & 255)) {
      int j = cur >> 8;
      s_cx = px[j]; s_cy = py[j]; s_cz = pz[j];
    }
    __syncthreads();
    float cx = s_cx, cy = s_cy, cz = s_cz;
    if (t == 0) {
      float* o = samp_xyz + ((size_t)b * NSAMP + it) * 3;
      o[0] = cx; o[1] = cy; o[2] = cz;
    }

    float bv = -1.0f;
    int   bi = 0;
#pragma unroll
    for (int j = 0; j < 16; ++j) {
      int n = j * 256 + t;
      float dx = px[j] - cx, dy = py[j] - cy, dz = pz[j] - cz;
      float d = dx * dx + dy * dy + dz * dz;
      md[j] = fminf(md[j], d);
      if (md[j] > bv) { bv = md[j]; bi = n; }   // ascending n => first-max kept
    }
    s_val[t] = bv; s_idx[t] = bi;
    __syncthreads();
    // argmax tree reduction, lowest index wins ties (matches jnp.argmax)
    for (int off = 128; off > 0; off >>= 1) {
      if (t < off) {
        float v2 = s_val[t + off]; int i2 = s_idx[t + off];
        if (v2 > s_val[t] || (v2 == s_val[t] && i2 < s_idx[t])) {
          s_val[t] = v2; s_idx[t] = i2;
        }
      }
      __syncthreads();
    }
    cur = s_idx[0];
    __syncthreads();
  }
}

// ---------------------------------------------------------------------------
// Kernel B: transpose x [B,C,N] -> feats [B,N,C] so neighbor-feature rows are
// contiguous 512B lines for the gather kernel. LDS 32x33 tile.
// ---------------------------------------------------------------------------
__global__ void transpose_kernel(const float* __restrict__ x,
                                 float* __restrict__ feats) {
  __shared__ float tile[32][33];
  const int b  = blockIdx.z;
  const int nt = blockIdx.x;   // N/32
  const int ct = blockIdx.y;   // C/32
  const int tx = threadIdx.x, ty = threadIdx.y;
  const float* xb = x + (size_t)b * NCH * NPTS;
#pragma unroll
  for (int i = 0; i < 32; i += 8)
    tile[ty + i][tx] = xb[(size_t)(ct * 32 + ty + i) * NPTS + nt * 32 + tx];
  __syncthreads();
  float* fb = feats + (size_t)b * NPTS * NCH;
#pragma unroll
  for (int i = 0; i < 32; i += 8)
    fb[(size_t)(nt * 32 + ty + i) * NCH + ct * 32 + tx] = tile[tx][ty + i];
}

// ---------------------------------------------------------------------------
// Kernel C: KNN via V_WMMA_F32_16X16X4_F32.
// Rank score = |c|^2 - 2 s.c  (the constant |s|^2 cannot change ordering).
// A row m  = (-2sx, -2sy, -2sz, 1)     (16x4)
// B col n  = ( cx ,  cy ,  cz , |c|^2) (4x16)
// One block = 16 samples, 16 waves. WMMA phase fills a [16][516]-padded LDS
// score tile per 512-column chunk; selection phase keeps the running top-32
// one-candidate-per-lane (K == wave32) with ballot-prefiltered inserts.
// ---------------------------------------------------------------------------
__global__ void knn_kernel(const float* __restrict__ coords,
                           const float* __restrict__ samp_xyz,
                           int* __restrict__ knn_idx) {
  const int b     = blockIdx.y;
  const int stile = blockIdx.x;          // 64 tiles of 16 samples
  const int t     = threadIdx.x;         // 512 threads = 16 waves
  const int wave  = t >> 5;
  const int lane  = t & 31;
  const int sm    = lane & 15;

  __shared__ float sc[16 * 516];

  const float* cb = coords   + (size_t)b * NPTS * 3;
  const float* sp = samp_xyz + ((size_t)b * NSAMP + stile * 16) * 3;

  // A-matrix fragment (f32 16x4): lanes 0-15 hold K=0,1; lanes 16-31 hold K=2,3
  v2f A;
  if (lane < 16) {
    A.x = -2.0f * sp[sm * 3 + 0];
    A.y = -2.0f * sp[sm * 3 + 1];
  } else {
    A.x = -2.0f * sp[sm * 3 + 2];
    A.y = 1.0f;
  }

  float myv    = FLT_BIG;  // per-lane member of the running top-32
  int   myi    = 0;
  float curmax = FLT_BIG;  // worst of the current top-32 (wave-uniform)
  int   maxlane = 0;

  for (int chunk = 0; chunk < 8; ++chunk) {
    const int cbase = chunk * 512;

    // ---- WMMA phase: each wave produces two 16x16 score tiles ----
#pragma unroll
    for (int tt = 0; tt < 2; ++tt) {
      const int col0 = cbase + (wave * 2 + tt) * 16;
      const int col  = col0 + sm;
      float cx = cb[col * 3 + 0];
      float cy = cb[col * 3 + 1];
      float cz = cb[col * 3 + 2];
      v2f Bf;
      if (lane < 16) { Bf.x = cx; Bf.y = cy; }
      else           { Bf.x = cz; Bf.y = cx * cx + cy * cy + cz * cz; }
      v8f Cm = {};
      Cm = __builtin_amdgcn_wmma_f32_16x16x4_f32(
          /*neg_a=*/false, A, /*neg_b=*/false, Bf,
          /*c_mod=*/(short)0, Cm, /*reuse_a=*/false, /*reuse_b=*/false);
      // D layout: lanes 0-15 -> rows 0-7, lanes 16-31 -> rows 8-15
      const int rbase = (lane < 16) ? 0 : 8;
      const int lcol  = (col0 - cbase) + sm;
#pragma unroll
      for (int r = 0; r < 8; ++r)
        sc[(rbase + r) * 516 + lcol] = Cm[r];
    }
    __syncthreads();

    // ---- selection phase: wave w owns sample row w ----
    for (int j = 0; j < 16; ++j) {
      float cv = sc[wave * 516 + j * 32 + lane];
      unsigned long long mask = __ballot(cv < curmax);
      while (mask) {
        int l = __ffsll(mask) - 1;
        mask &= mask - 1;
        float d = __shfl(cv, l);
        if (d < curmax) {
          int gi = cbase + j * 32 + l;
          if (lane == maxlane) { myv = d; myi = gi; }
          float m = myv;
#pragma unroll
          for (int o = 16; o > 0; o >>= 1) m = fmaxf(m, __shfl_xor(m, o));
          curmax = m;
          unsigned long long bm = __ballot(myv == curmax);
          maxlane = __ffsll(bm) - 1;
        }
      }
    }
    __syncthreads();
  }

  knn_idx[((size_t)b * NSAMP + stile * 16 + wave) * KNN + lane] = myi;
}

// ---------------------------------------------------------------------------
// Kernel D: gather neighbor features + max-pool. One 128-thread block per
// sample; 32 contiguous 512B rows, coalesced across channels.
// ---------------------------------------------------------------------------
__global__ void gather_pool_kernel(const float* __restrict__ feats,
                                   const int* __restrict__ knn_idx,
                                   float* __restrict__ out) {
  const int s = blockIdx.x;
  const int b = blockIdx.y;
  const int c = threadIdx.x;   // 128 channels

  __shared__ int si[KNN];
  const int* kp = knn_idx + ((size_t)b * NSAMP + s) * KNN;
  if (c < KNN) si[c] = kp[c];
  __syncthreads();

  const float* fb = feats + (size_t)b * NPTS * NCH;
  float m = -FLT_BIG;
#pragma unroll 4
  for (int k = 0; k < KNN; ++k)
    m = fmaxf(m, fb[(size_t)si[k] * NCH + c]);

  out[((size_t)b * NSAMP + s) * NCH + c] = m;
}

// ---------------------------------------------------------------------------
extern "C" void kernel_launch(void* const* d_in, const int* in_sizes, int n_in,
                              void* d_out, int out_size, void* d_ws, size_t ws_size,
                              hipStream_t stream) {
  const float* x      = (const float*)d_in[0];   // [16,128,4096]
  const float* coords = (const float*)d_in[1];   // [16,4096,3]
  float* out = (float*)d_out;                    // [16, 1024*128]

  // workspace layout
  char* ws = (char*)d_ws;
  float* feats    = (float*)ws;                                  // B*N*C f32 = 32 MB
  ws += (size_t)NBATCH * NPTS * NCH * sizeof(float);
  float* samp_xyz = (float*)ws;                                  // B*S*3 f32
  ws += (size_t)NBATCH * NSAMP * 3 * sizeof(float);
  int* knn_idx    = (int*)ws;                                    // B*S*K i32 = 2 MB

  fps_kernel<<<NBATCH, 256, 0, stream>>>(coords, samp_xyz);
  transpose_kernel<<<dim3(NPTS / 32, NCH / 32, NBATCH), dim3(32, 8), 0, stream>>>(x, feats);
  knn_kernel<<<dim3(NSAMP / 16, NBATCH), 512, 0, stream>>>(coords, samp_xyz, knn_idx);
  gather_pool_kernel<<<dim3(NSAMP, NBATCH), NCH, 0, stream>>>(feats, knn_idx, out);
}